// AttentionPooling_39762807227086
// MI455X (gfx1250) — compile-verified
//
#include <hip/hip_runtime.h>
#include <math.h>

#define HID 128

typedef float v2f __attribute__((ext_vector_type(2)));
typedef float v8f __attribute__((ext_vector_type(8)));
typedef int   v4i __attribute__((ext_vector_type(4)));

#define GLOBAL_AS __attribute__((address_space(1)))
#define LDS_AS    __attribute__((address_space(3)))

// ---- gfx1250 async global->LDS path (guarded; fallback = float4 round-trip)
#if defined(__has_builtin)
#if __has_builtin(__builtin_amdgcn_global_load_async_to_lds_b128)
#define HAVE_ASYNC_LDS 1
#endif
#if __has_builtin(__builtin_amdgcn_s_wait_asynccnt)
#define HAVE_WAIT_ASYNC 1
#endif
#endif

__device__ __forceinline__ void wait_asynccnt0() {
#if defined(HAVE_WAIT_ASYNC)
    __builtin_amdgcn_s_wait_asynccnt(0);
#else
    asm volatile("s_wait_asynccnt 0" ::: "memory");
#endif
}

// Branchless tanh via native exp/rcp: exact at both saturation ends,
// ~1e-7 abs error, ~5 VALU ops vs ~40 branchy ops for libm tanhf.
__device__ __forceinline__ float fast_tanh(float v) {
    float e = __expf(2.f * v);                 // v_exp_f32 (+scale)
    return 1.f - 2.f * __builtin_amdgcn_rcpf(e + 1.f);
}

// ---------------------------------------------------------------------------
// Kernel 1: logits[n] = tanh(x[n,:] @ W1 + b1) @ W2 + b2
// Block = 128 threads (4 waves); block covers 64 rows, each wave 16 rows.
// x tile staged in LDS (row pad 132 floats -> conflict-free A reads).
// GEMM via V_WMMA_F32_16X16X4_F32 (fp32-exact), K-loop 32 steps, 8 col tiles.
// ---------------------------------------------------------------------------
__global__ __launch_bounds__(128) void ap_logits_kernel(
    const float* __restrict__ x, const float* __restrict__ W1,
    const float* __restrict__ b1, const float* __restrict__ W2,
    const float* __restrict__ b2, float* __restrict__ logits, int n)
{
    __shared__ float xs[64 * 132];
    const int tid  = threadIdx.x;
    const int lane = tid & 31;
    const int wave = tid >> 5;
    const long long rowBase = (long long)blockIdx.x * 64;

    // Stage 64x128 x-tile into LDS.
#if defined(HAVE_ASYNC_LDS)
    for (int i = tid; i < 64 * 32; i += 128) {
        const int r  = i >> 5;
        const int cv = (i & 31) << 2;
        if (rowBase + r < n) {
            __builtin_amdgcn_global_load_async_to_lds_b128(
                (GLOBAL_AS v4i*)(x + (rowBase + r) * HID + cv),
                (LDS_AS v4i*)(&xs[r * 132 + cv]),
                0, 0);
        } else {
            *(float4*)(&xs[r * 132 + cv]) = make_float4(0.f, 0.f, 0.f, 0.f);
        }
    }
    wait_asynccnt0();
#else
    for (int i = tid; i < 64 * 32; i += 128) {
        const int r  = i >> 5;
        const int cv = (i & 31) << 2;
        float4 v = make_float4(0.f, 0.f, 0.f, 0.f);
        if (rowBase + r < n) v = *(const float4*)(x + (rowBase + r) * HID + cv);
        *(float4*)(&xs[r * 132 + cv]) = v;
    }
#endif
    __syncthreads();

    const int half = lane >> 4;   // 0: lanes 0-15, 1: lanes 16-31
    const int l16  = lane & 15;
    const float b2v = b2[0];

    float lp[8];
#pragma unroll
    for (int j = 0; j < 8; ++j) lp[j] = 0.f;

    // A-fragment base: lane l16 owns row (wave*16+l16); half selects K pair.
    const float* ap = &xs[(wave * 16 + l16) * 132 + half * 2];

    for (int c = 0; c < 8; ++c) {
        v8f acc = {0.f, 0.f, 0.f, 0.f, 0.f, 0.f, 0.f, 0.f};
        // B-fragment: lanes 0-15 rows k0,k0+1 ; lanes 16-31 rows k0+2,k0+3.
        const float* bp = W1 + half * 2 * HID + c * 16 + l16;
#pragma unroll 4
        for (int k0 = 0; k0 < HID; k0 += 4) {
            v2f a = *(const v2f*)(ap + k0);
            v2f b;
            b.x = bp[k0 * HID];
            b.y = bp[(k0 + 1) * HID];
            acc = __builtin_amdgcn_wmma_f32_16x16x4_f32(
                false, a, false, b, (short)0, acc, false, false);
        }
        const float bb  = b1[c * 16 + l16];
        const float w2v = W2[c * 16 + l16];
#pragma unroll
        for (int j = 0; j < 8; ++j) {
            lp[j] += fast_tanh(acc[j] + bb) * w2v;   // fused tanh + dot(W2)
        }
    }

    // Reduce partial logits across the 16 lanes of each half (D layout:
    // VGPR j / lanes 0-15 -> row j ; lanes 16-31 -> row j+8).
#pragma unroll
    for (int j = 0; j < 8; ++j) {
        float v = lp[j];
        v += __shfl_xor(v, 1, 16);
        v += __shfl_xor(v, 2, 16);
        v += __shfl_xor(v, 4, 16);
        v += __shfl_xor(v, 8, 16);
        lp[j] = v;
    }
    if (l16 == 0) {
        const long long r0 = rowBase + wave * 16 + half * 8;
#pragma unroll
        for (int j = 0; j < 8; ++j)
            if (r0 + j < n) logits[r0 + j] = lp[j] + b2v;
    }
}

// ---------------------------------------------------------------------------
// Kernel 2: per-graph softmax over logits + weighted pooling of x.
// One 128-thread block per graph. batch is sorted -> binary search the
// segment. All reductions are fixed-tree / sequential => deterministic.
// ---------------------------------------------------------------------------
__global__ __launch_bounds__(128) void ap_pool_kernel(
    const float* __restrict__ x, const int* __restrict__ batch,
    const float* __restrict__ logits, float* __restrict__ out, int n)
{
    const int g = blockIdx.x;
    const int t = threadIdx.x;

    // lower_bound(batch, g) and lower_bound(batch, g+1)
    int lo = 0, hi = n;
    while (lo < hi) { int m = (lo + hi) >> 1; if (batch[m] < g) lo = m + 1; else hi = m; }
    const int start = lo;
    hi = n;
    while (lo < hi) { int m = (lo + hi) >> 1; if (batch[m] < g + 1) lo = m + 1; else hi = m; }
    const int end = lo;

    __shared__ float red[128];
    __shared__ float wbuf[1024];

    // segment max (fixed tree -> deterministic)
    float mx = -INFINITY;
    for (int i = start + t; i < end; i += 128) mx = fmaxf(mx, logits[i]);
    red[t] = mx; __syncthreads();
#pragma unroll
    for (int s = 64; s > 0; s >>= 1) {
        if (t < s) red[t] = fmaxf(red[t], red[t + s]);
        __syncthreads();
    }
    mx = red[0]; __syncthreads();

    // segment sum of exp(logit - max)
    float sm = 0.f;
    for (int i = start + t; i < end; i += 128) sm += expf(logits[i] - mx);
    red[t] = sm; __syncthreads();
#pragma unroll
    for (int s = 64; s > 0; s >>= 1) {
        if (t < s) red[t] += red[t + s];
        __syncthreads();
    }
    const float total = red[0];
    const float inv = (total > 0.f) ? (1.f / total) : 0.f;
    __syncthreads();

    // weighted pooling: thread t owns feature t, sequential over nodes
    float acc = 0.f;
    for (int cs = start; cs < end; cs += 1024) {
        const int clen = min(1024, end - cs);
        for (int i = t; i < clen; i += 128)
            wbuf[i] = expf(logits[cs + i] - mx) * inv;
        __syncthreads();
        for (int i = 0; i < clen; ++i)
            acc += wbuf[i] * x[(long long)(cs + i) * HID + t];
        __syncthreads();
    }
    out[(long long)g * HID + t] = acc;
}

// ---------------------------------------------------------------------------
extern "C" void kernel_launch(void* const* d_in, const int* in_sizes, int n_in,
                              void* d_out, int out_size, void* d_ws, size_t ws_size,
                              hipStream_t stream)
{
    const float* x     = (const float*)d_in[0];
    const int*   batch = (const int*)d_in[1];
    const float* W1    = (const float*)d_in[2];
    const float* b1    = (const float*)d_in[3];
    const float* W2    = (const float*)d_in[4];
    const float* b2    = (const float*)d_in[5];
    float* out = (float*)d_out;

    const int n        = in_sizes[1];          // N_NODES
    const int n_graphs = out_size / HID;       // N_GRAPHS

    float* logits = (float*)d_ws;              // N floats of scratch

    const int gridA = (n + 63) / 64;
    ap_logits_kernel<<<gridA, 128, 0, stream>>>(x, W1, b1, W2, b2, logits, n);
    ap_pool_kernel<<<n_graphs, 128, 0, stream>>>(x, batch, logits, out, n);
}